// TopKEdgePooling_53188874993833
// MI455X (gfx1250) — compile-verified
//
#include <hip/hip_runtime.h>
#include <stdint.h>

// Problem constants (from the reference)
#define B_    64
#define NPG_  3125
#define F_    128
#define N_    200000
#define EPG_  50000
#define E_    3200000
#define HID_  128
#define KSEL_ 25000      // ceil(0.5 * EPG)
#define PROP_ 12

typedef float v2f __attribute__((ext_vector_type(2)));
typedef float v8f __attribute__((ext_vector_type(8)));
typedef unsigned int v4u __attribute__((ext_vector_type(4)));
typedef int v4i __attribute__((ext_vector_type(4)));
typedef int v8i __attribute__((ext_vector_type(8)));

// ---------------------------------------------------------------------------
// Composite 48-bit key: [47:16] = order-preserving float bits, [15:0] = index
// tiebreak (smaller original index -> larger key), so the k-th largest key is
// unique and "key >= kth" selects exactly KSEL_ edges, matching the stable
// lexsort((-score, seg)) semantics of the reference.
// ---------------------------------------------------------------------------
__device__ __forceinline__ uint64_t make_key(float f, int lidx) {
    uint32_t u = __float_as_uint(f);
    u = (u & 0x80000000u) ? ~u : (u | 0x80000000u);   // ascending-order map
    return ((uint64_t)u << 16) | (uint32_t)(EPG_ - 1 - lidx);
}

// ---------------------------------------------------------------------------
// Init: zero histograms & counts, reset per-graph state, lab[i] = i
// ---------------------------------------------------------------------------
__global__ __launch_bounds__(256) void k_init(uint32_t* hist, uint64_t* pref,
                                              int* krem, int* maxsz, int* bestl,
                                              int* cnt, int* lab) {
    int i = blockIdx.x * 256 + threadIdx.x;
    if (i < B_ * 256) hist[i] = 0u;
    if (i < B_) { pref[i] = 0ull; krem[i] = KSEL_; maxsz[i] = 0; bestl[i] = N_; }
    if (i < N_) { cnt[i] = 0; lab[i] = i; }
}

// ---------------------------------------------------------------------------
// Edge MLP via V_WMMA_F32_16X16X4_F32, with the block's edge_attr tile
// (128 edges x 2 f32 = 1 KB) staged into LDS by the Tensor Data Mover.
//
// D# descriptor (groups 0/1, 1-D tile): data_size=4B, tile_dim0=256,
// tensor_dim0=256, tensor_dim0_stride=256, tile_dim1=tensor_dim1=1.
// One wave issues tensor_load_to_lds, waits on TENSORcnt, then the block
// barrier publishes LDS to all 8 waves.
//
// WMMA: A (16x4): lanes 0-15 = K0,K1 from edge_attr; lanes 16-31 = pad.
//       B (4x16): lanes 0-15 = w1 rows 0,1 for cols t*16+l16; lanes 16-31 pad.
//       D (16x16 f32): VGPR v holds rows v / v+8, col = lane%16.
// Second layer = relu(D + b1) * w2, butterfly-reduced over 16 lanes.
// ---------------------------------------------------------------------------
__global__ __launch_bounds__(256) void k_edge_mlp(const float* __restrict__ attr,
                                                  const float* __restrict__ w1,
                                                  const float* __restrict__ b1v,
                                                  const float* __restrict__ w2,
                                                  const float* __restrict__ b2v,
                                                  float* __restrict__ score_raw) {
    __shared__ float sattr[256];        // 128 edges x 2 attrs

    const int tid  = threadIdx.x;
    const int lane = tid & 31;
    const int wib  = tid >> 5;          // wave-in-block 0..7
    const int half = lane >> 4;         // 0: lanes 0-15, 1: lanes 16-31
    const int l16  = lane & 15;
    const int eblk = blockIdx.x * 128;  // first edge of this block

    if (wib == 0) {
        const uint64_t ga = (uint64_t)(uintptr_t)(attr + 2 * (size_t)eblk);
        const uint32_t la = (uint32_t)(uintptr_t)(&sattr[0]);  // LDS byte offset
        v4u g0;
        g0.x = 1u;                                  // count=1, load, no gather
        g0.y = la;                                  // lds_addr
        g0.z = (uint32_t)ga;                        // global_addr[31:0]
        g0.w = (uint32_t)(ga >> 32) | (2u << 30);   // global_addr[56:32] | type=2
        v8i g1;
        g1.s0 = 0x00020000;                         // data_size = 4 bytes
        g1.s1 = (int)(256u << 16);                  // tensor_dim0 = 256 (lo16)
        g1.s2 = (int)(1u << 16);                    // tensor_dim0 hi, tensor_dim1 = 1
        g1.s3 = (int)(256u << 16);                  // tensor_dim1 hi, tile_dim0 = 256
        g1.s4 = 1;                                  // tile_dim1 = 1, tile_dim2 = 0
        g1.s5 = 256;                                // tensor_dim0_stride = 256
        g1.s6 = 0;
        g1.s7 = 0;
        v4i g2 = {0, 0, 0, 0};
        v4i g3 = {0, 0, 0, 0};
        v8i g4 = {0, 0, 0, 0, 0, 0, 0, 0};
        __builtin_amdgcn_tensor_load_to_lds(g0, g1, g2, g3, g4, 0);
        __builtin_amdgcn_s_wait_tensorcnt(0);
    }
    __syncthreads();

    // A fragment from LDS (lanes 16-31 read a valid mirror, then zeroed)
    const float2 t = *reinterpret_cast<const float2*>(&sattr[2 * (wib * 16 + l16)]);
    v2f a;
    a.x = half ? 0.0f : t.x;
    a.y = half ? 0.0f : t.y;

    // Hoist all weights (unconditional loads, valid for every lane)
    float w1a[8], w1b[8], b1r[8], w2r[8];
#pragma unroll
    for (int t8 = 0; t8 < 8; ++t8) {
        const int n = t8 * 16 + l16;
        w1a[t8] = w1[n];
        w1b[t8] = w1[HID_ + n];
        b1r[t8] = b1v[n];
        w2r[t8] = w2[n];
    }

    float acc[8];
#pragma unroll
    for (int v = 0; v < 8; ++v) acc[v] = 0.0f;

#pragma unroll
    for (int t8 = 0; t8 < 8; ++t8) {
        v2f b;
        b.x = half ? 0.0f : w1a[t8];
        b.y = half ? 0.0f : w1b[t8];
        v8f c = {0.f, 0.f, 0.f, 0.f, 0.f, 0.f, 0.f, 0.f};
        v8f d = __builtin_amdgcn_wmma_f32_16x16x4_f32(
            /*neg_a=*/false, a, /*neg_b=*/false, b,
            /*c_mod=*/(short)0, c, /*reuse_a=*/false, /*reuse_b=*/false);
        const float bn = b1r[t8];
        const float wn = w2r[t8];
#pragma unroll
        for (int v = 0; v < 8; ++v) {
            float h = d[v] + bn;
            h = h > 0.0f ? h : 0.0f;     // ReLU
            acc[v] += h * wn;            // partial dot with w2
        }
    }

    // Reduce over the 16 lanes of each half (xor<16 never crosses halves)
#pragma unroll
    for (int v = 0; v < 8; ++v) {
#pragma unroll
        for (int off = 1; off < 16; off <<= 1)
            acc[v] += __shfl_xor(acc[v], off, 32);
    }

    const int e0 = eblk + wib * 16;
    const float bias2 = b2v[0];
#pragma unroll
    for (int v = 0; v < 8; ++v)
        if (l16 == v) score_raw[e0 + v + 8 * half] = acc[v] + bias2;
}

// ---------------------------------------------------------------------------
// Per-graph softmax (3 passes over the score region in d_out)
// ---------------------------------------------------------------------------
__global__ __launch_bounds__(256) void k_seg_max(const float* __restrict__ score,
                                                 float* __restrict__ gmax) {
    const int g = blockIdx.x;
    const float* s = score + (size_t)g * EPG_;
    float mx = -3.4e38f;
    for (int e = threadIdx.x; e < EPG_; e += 256) mx = fmaxf(mx, s[e]);
    __shared__ float red[256];
    red[threadIdx.x] = mx; __syncthreads();
    for (int o = 128; o > 0; o >>= 1) {
        if (threadIdx.x < o) red[threadIdx.x] = fmaxf(red[threadIdx.x], red[threadIdx.x + o]);
        __syncthreads();
    }
    if (threadIdx.x == 0) gmax[g] = red[0];
}

__global__ __launch_bounds__(256) void k_exp_sum(float* __restrict__ score,
                                                 const float* __restrict__ gmax,
                                                 float* __restrict__ gsum) {
    const int g = blockIdx.x;
    const float m = gmax[g];
    float* s = score + (size_t)g * EPG_;
    float acc = 0.0f;
    for (int e = threadIdx.x; e < EPG_; e += 256) {
        const float ex = expf(s[e] - m);
        s[e] = ex;
        acc += ex;
    }
    __shared__ float red[256];
    red[threadIdx.x] = acc; __syncthreads();
    for (int o = 128; o > 0; o >>= 1) {
        if (threadIdx.x < o) red[threadIdx.x] += red[threadIdx.x + o];
        __syncthreads();
    }
    if (threadIdx.x == 0) gsum[g] = red[0];
}

__global__ __launch_bounds__(256) void k_norm(float* __restrict__ score,
                                              const float* __restrict__ gsum) {
    const int e = blockIdx.x * 256 + threadIdx.x;
    const int g = e / EPG_;
    score[e] = score[e] / (gsum[g] + 1e-16f);
}

// ---------------------------------------------------------------------------
// Radix select (6 passes x 8 bits over the 48-bit key) -> exact k-th key
// ---------------------------------------------------------------------------
__global__ __launch_bounds__(256) void k_hist(const float* __restrict__ score,
                                              const uint64_t* __restrict__ pref,
                                              uint32_t* __restrict__ hist,
                                              int shift) {
    const int g = blockIdx.y;
    __shared__ unsigned int lh[256];
    lh[threadIdx.x] = 0u; __syncthreads();
    const uint64_t p = pref[g];
    const float* s = score + (size_t)g * EPG_;
    const int stride = gridDim.x * 256;
    for (int e = blockIdx.x * 256 + threadIdx.x; e < EPG_; e += stride) {
        __builtin_prefetch(s + e + stride, 0, 0);   // global_prefetch_b8
        const uint64_t key = make_key(s[e], e);
        if ((key >> (shift + 8)) == p)
            atomicAdd(&lh[(unsigned)(key >> shift) & 255u], 1u);
    }
    __syncthreads();
    if (lh[threadIdx.x]) atomicAdd(&hist[g * 256 + threadIdx.x], lh[threadIdx.x]);
}

__global__ __launch_bounds__(256) void k_select(uint32_t* __restrict__ hist,
                                                uint64_t* __restrict__ pref,
                                                int* __restrict__ krem) {
    const int g = blockIdx.x;
    __shared__ unsigned int lh[256];
    lh[threadIdx.x] = hist[g * 256 + threadIdx.x];
    __syncthreads();
    if (threadIdx.x == 0) {
        int k = krem[g];
        unsigned int cum = 0; int bin = 0;
        for (int v = 255; v >= 0; --v) {
            const unsigned int c = lh[v];
            if (cum + c >= (unsigned int)k) { bin = v; break; }
            cum += c;
        }
        krem[g] = k - (int)cum;
        pref[g] = (pref[g] << 8) | (unsigned int)bin;
    }
    hist[g * 256 + threadIdx.x] = 0u;   // ready for next pass
}

__global__ __launch_bounds__(256) void k_keep(const float* __restrict__ score,
                                              const uint64_t* __restrict__ kth,
                                              uint8_t* __restrict__ keep) {
    const int e = blockIdx.x * 256 + threadIdx.x;
    const int g = e / EPG_;
    const int le = e - g * EPG_;
    keep[e] = (make_key(score[e], le) >= kth[g]) ? 1 : 0;
}

// ---------------------------------------------------------------------------
// Connected components: min-label propagation + pointer jumping
// ---------------------------------------------------------------------------
__global__ __launch_bounds__(256) void k_edge_min(const int* __restrict__ src,
                                                  const int* __restrict__ dst,
                                                  const uint8_t* __restrict__ keep,
                                                  int* __restrict__ lab) {
    const int e = blockIdx.x * 256 + threadIdx.x;
    if (!keep[e]) return;                 // dropped edges -> (0,0): inert
    const int s = src[e], d = dst[e];
    const int ls = lab[s], ld = lab[d];
    const int m = ls < ld ? ls : ld;
    if (ls > m) atomicMin(&lab[s], m);
    if (ld > m) atomicMin(&lab[d], m);
}

__global__ __launch_bounds__(256) void k_jump(const int* __restrict__ lab,
                                              int* __restrict__ out) {
    const int i = blockIdx.x * 256 + threadIdx.x;
    if (i < N_) out[i] = lab[lab[i]];
}

__global__ __launch_bounds__(256) void k_count(const int* __restrict__ lab,
                                               int* __restrict__ cnt) {
    const int i = blockIdx.x * 256 + threadIdx.x;
    if (i < N_) atomicAdd(&cnt[lab[i]], 1);
}

__global__ __launch_bounds__(256) void k_maxsz(const int* __restrict__ lab,
                                               const int* __restrict__ cnt,
                                               int* __restrict__ maxsz) {
    const int i = blockIdx.x * 256 + threadIdx.x;
    if (i < N_) atomicMax(&maxsz[i / NPG_], cnt[lab[i]]);
}

__global__ __launch_bounds__(256) void k_bestl(const int* __restrict__ lab,
                                               const int* __restrict__ cnt,
                                               const int* __restrict__ maxsz,
                                               int* __restrict__ bestl) {
    const int i = blockIdx.x * 256 + threadIdx.x;
    if (i < N_) {
        const int g = i / NPG_;
        const int l = lab[i];
        if (cnt[l] == maxsz[g]) atomicMin(&bestl[g], l);
    }
}

// ---------------------------------------------------------------------------
// Output masking (vectorized)
// ---------------------------------------------------------------------------
__global__ __launch_bounds__(256) void k_xout(const float* __restrict__ x,
                                              const int* __restrict__ lab,
                                              const int* __restrict__ bestl,
                                              float* __restrict__ out) {
    const int idx = blockIdx.x * 256 + threadIdx.x;   // float4 index
    const int node = idx >> 5;                        // 32 float4 per node
    const float m = (lab[node] == bestl[node / NPG_]) ? 1.0f : 0.0f;
    const float4 v = reinterpret_cast<const float4*>(x)[idx];
    float4 o; o.x = v.x * m; o.y = v.y * m; o.z = v.z * m; o.w = v.w * m;
    reinterpret_cast<float4*>(out)[idx] = o;
}

__global__ __launch_bounds__(256) void k_eout(const float* __restrict__ attr,
                                              const int* __restrict__ src,
                                              const int* __restrict__ dst,
                                              const uint8_t* __restrict__ keep,
                                              const int* __restrict__ lab,
                                              const int* __restrict__ bestl,
                                              float* __restrict__ out) {
    const int e = blockIdx.x * 256 + threadIdx.x;
    const int g = e / EPG_;
    const int bl = bestl[g];
    const float m = (keep[e] && lab[src[e]] == bl && lab[dst[e]] == bl) ? 1.0f : 0.0f;
    const float2 v = reinterpret_cast<const float2*>(attr)[e];
    float2 o; o.x = v.x * m; o.y = v.y * m;
    reinterpret_cast<float2*>(out)[e] = o;
}

// ---------------------------------------------------------------------------
extern "C" void kernel_launch(void* const* d_in, const int* in_sizes, int n_in,
                              void* d_out, int out_size, void* d_ws, size_t ws_size,
                              hipStream_t stream) {
    (void)in_sizes; (void)n_in; (void)out_size; (void)ws_size;

    const float* x     = (const float*)d_in[0];
    const int*   eidx  = (const int*)d_in[1];
    const float* eattr = (const float*)d_in[2];
    // d_in[3]=batch, d_in[4]=edge_batch: recomputed on-device (i/NPG, e/EPG)
    const float* w1 = (const float*)d_in[5];
    const float* b1 = (const float*)d_in[6];
    const float* w2 = (const float*)d_in[7];
    const float* b2 = (const float*)d_in[8];

    float* x_out = (float*)d_out;                                  // N*F
    float* e_out = (float*)d_out + (size_t)N_ * F_;                // E*2
    float* score = (float*)d_out + (size_t)N_ * F_ + (size_t)E_*2; // E

    char* ws = (char*)d_ws;   // ~5.7 MB total, 256B-aligned regions
    int*      labA  = (int*)     (ws + 0);
    int*      labB  = (int*)     (ws + 800000);
    int*      cnt   = (int*)     (ws + 1600000);
    uint8_t*  keep  = (uint8_t*) (ws + 2400000);
    uint32_t* hist  = (uint32_t*)(ws + 5600000);
    uint64_t* pref  = (uint64_t*)(ws + 5665536);
    int*      krem  = (int*)     (ws + 5666048);
    float*    gmax  = (float*)   (ws + 5666560);
    float*    gsum  = (float*)   (ws + 5666816);
    int*      maxsz = (int*)     (ws + 5667072);
    int*      bestl = (int*)     (ws + 5667328);

    const int* src = eidx;
    const int* dst = eidx + E_;

    const int nb_n = (N_ + 255) / 256;

    k_init<<<nb_n, 256, 0, stream>>>(hist, pref, krem, maxsz, bestl, cnt, labA);

    // 1) Edge-score MLP (TDM tile stage + WMMA f32 16x16x4)
    k_edge_mlp<<<E_ / 128, 256, 0, stream>>>(eattr, w1, b1, w2, b2, score);

    // 2) Per-graph softmax (in place in score region)
    k_seg_max<<<B_, 256, 0, stream>>>(score, gmax);
    k_exp_sum<<<B_, 256, 0, stream>>>(score, gmax, gsum);
    k_norm<<<E_ / 256, 256, 0, stream>>>(score, gsum);

    // 3) Exact per-graph top-KSEL_ via 48-bit radix select
    for (int p = 0; p < 6; ++p) {
        const int shift = 40 - 8 * p;
        k_hist<<<dim3(8, B_), 256, 0, stream>>>(score, pref, hist, shift);
        k_select<<<B_, 256, 0, stream>>>(hist, pref, krem);
    }
    k_keep<<<E_ / 256, 256, 0, stream>>>(score, pref, keep);

    // 4) Components: 12 x (edge min-scatter + pointer jump)
    int* cur = labA; int* nxt = labB;
    for (int it = 0; it < PROP_; ++it) {
        k_edge_min<<<E_ / 256, 256, 0, stream>>>(src, dst, keep, cur);
        k_jump<<<nb_n, 256, 0, stream>>>(cur, nxt);
        int* t = cur; cur = nxt; nxt = t;
    }

    // 5) Component sizes -> per-graph maximal component
    k_count<<<nb_n, 256, 0, stream>>>(cur, cnt);
    k_maxsz<<<nb_n, 256, 0, stream>>>(cur, cnt, maxsz);
    k_bestl<<<nb_n, 256, 0, stream>>>(cur, cnt, maxsz, bestl);

    // 6) Masked outputs
    k_xout<<<(N_ * F_ / 4) / 256, 256, 0, stream>>>(x, cur, bestl, x_out);
    k_eout<<<E_ / 256, 256, 0, stream>>>(eattr, src, dst, keep, cur, bestl, e_out);
}